// ChamferDistance_2542620639339
// MI455X (gfx1250) — compile-verified
//
#include <hip/hip_runtime.h>

typedef float v2f __attribute__((ext_vector_type(2)));
typedef float v8f __attribute__((ext_vector_type(8)));

#define BLOCK       256   // 8 waves (wave32)
#define ROWS_PER_WG 128   // 8 waves * 16 rows
#define CHUNK       4096  // columns staged per LDS pass (4096 * 16B = 64KB)

__global__ __launch_bounds__(64) void chamfer_init_kernel(float* out) {
    if (threadIdx.x == 0) out[0] = 0.0f;
}

// Direct v_min_num_f32 without fminf's sNaN-canonicalization (inputs never NaN).
__device__ __forceinline__ void min_acc(float& m, float v) {
    asm("v_min_num_f32 %0, %0, %1" : "+v"(m) : "v"(v));
}

// One direction of Chamfer distance:
//   for each "row" point a (from rowPts), min over all "col" points b (from colPts)
//   of ||a-b||^2 = ||a||^2 + ||b||^2 - 2 a.b, accumulated (scaled) into out[0].
// WMMA encoding: A-row = [-2ax, -2ay, -2az, 1], B-col = [bx, by, bz, ||b||^2]
//   => A.B = ||b||^2 - 2 a.b ; add row-constant ||a||^2 after the min-reduction.
__global__ __launch_bounds__(BLOCK) void chamfer_dir_kernel(
    const float* __restrict__ rowPts,   // [B, nRows, 3]
    const float* __restrict__ colPts,   // [B, nCols, 3]
    int nRows, int nCols, float scale,
    float* __restrict__ out)
{
    __shared__ float4 tile[CHUNK];      // per column: {x, y, z, ||b||^2}

    const int b    = blockIdx.y;
    const int lane = threadIdx.x & 31;
    const int wave = threadIdx.x >> 5;
    const int half = lane >> 4;         // 0: lanes 0-15, 1: lanes 16-31
    const int l16  = lane & 15;

    const float* rp = rowPts + (size_t)b * nRows * 3;
    const float* cp = colPts + (size_t)b * nCols * 3;

    // ---- A fragment: this wave's 16-row strip (16x4 f32, per ISA layout:
    //      lanes 0-15 hold K=0,1 ; lanes 16-31 hold K=2,3 for row M = lane%16) ----
    const int rowBase = blockIdx.x * ROWS_PER_WG + wave * 16;
    int myRow = rowBase + l16;
    int rc = (myRow < nRows) ? myRow : (nRows - 1);
    float ax = rp[3 * rc + 0];
    float ay = rp[3 * rc + 1];
    float az = rp[3 * rc + 2];
    float sqMine = ax * ax + ay * ay + az * az;

    v2f A;
    A.x = half ? (-2.0f * az) : (-2.0f * ax);   // K=0 / K=2
    A.y = half ? 1.0f         : (-2.0f * ay);   // K=1 / K=3

    // ||a||^2 for the rows this lane's C fragment covers (M = 8*half + v),
    // fetched from the lane that loaded that row (lane 8*half+v holds row rowBase+8*half+v).
    float sqRow[8];
    #pragma unroll
    for (int v = 0; v < 8; ++v)
        sqRow[v] = __shfl(sqMine, 8 * half + v, 32);

    float rmin[8];
    #pragma unroll
    for (int v = 0; v < 8; ++v) rmin[v] = __builtin_inff();

    v8f czero = {};

    for (int cc = 0; cc < nCols; cc += CHUNK) {
        __syncthreads();   // previous chunk fully consumed before refill
        // ---- cooperative stage of column points into LDS ----
        for (int i = threadIdx.x; i < CHUNK; i += BLOCK) {
            int col = cc + i;
            int c = (col < nCols) ? col : (nCols - 1);
            float x = cp[3 * c + 0];
            float y = cp[3 * c + 1];
            float z = cp[3 * c + 2];
            float sq = (col < nCols) ? (x * x + y * y + z * z) : __builtin_inff();
            tile[i] = make_float4(x, y, z, sq);
        }
        __syncthreads();

        int lim = nCols - cc;
        if (lim > CHUNK) lim = CHUNK;

        // per-lane LDS base: half 0 reads {x,y} (K=0,1), half 1 reads {z,sq} (K=2,3)
        const float* lds2 = ((const float*)tile) + l16 * 4 + half * 2;

        if (lim == CHUNK) {
            // compile-time trip count -> unrolled, loads hoisted across WMMAs
            #pragma unroll 4
            for (int c0 = 0; c0 < CHUNK; c0 += 16) {
                v2f Bm;
                Bm.x = lds2[c0 * 4 + 0];
                Bm.y = lds2[c0 * 4 + 1];
                v8f acc = __builtin_amdgcn_wmma_f32_16x16x4_f32(
                    false, A, false, Bm, (short)0, czero, false, false);
                #pragma unroll
                for (int v = 0; v < 8; ++v) min_acc(rmin[v], acc[v]);
            }
        } else {
            for (int c0 = 0; c0 < lim; c0 += 16) {
                v2f Bm;
                Bm.x = lds2[c0 * 4 + 0];
                Bm.y = lds2[c0 * 4 + 1];
                v8f acc = __builtin_amdgcn_wmma_f32_16x16x4_f32(
                    false, A, false, Bm, (short)0, czero, false, false);
                #pragma unroll
                for (int v = 0; v < 8; ++v) min_acc(rmin[v], acc[v]);
            }
        }
    }

    // add row-constant ||a||^2 to get true squared distances
    #pragma unroll
    for (int v = 0; v < 8; ++v) rmin[v] += sqRow[v];

    // min across the 16 lanes of the half (C layout: lane index = column N)
    #pragma unroll
    for (int v = 0; v < 8; ++v) {
        float m = rmin[v];
        m = fminf(m, __shfl_xor(m, 1, 32));
        m = fminf(m, __shfl_xor(m, 2, 32));
        m = fminf(m, __shfl_xor(m, 4, 32));
        m = fminf(m, __shfl_xor(m, 8, 32));
        rmin[v] = m;
    }

    // one lane per half contributes its 8 row-mins (scaled) to the global mean
    if (l16 == 0) {
        float s = 0.0f;
        #pragma unroll
        for (int v = 0; v < 8; ++v) {
            int r = rowBase + 8 * half + v;
            s += (r < nRows) ? rmin[v] : 0.0f;
        }
        atomicAdd(out, s * scale);
    }
}

extern "C" void kernel_launch(void* const* d_in, const int* in_sizes, int n_in,
                              void* d_out, int out_size, void* d_ws, size_t ws_size,
                              hipStream_t stream) {
    const float* xyz1 = (const float*)d_in[0];
    const float* xyz2 = (const float*)d_in[1];
    float* out = (float*)d_out;

    const int Bsz = 4, D = 3;
    int N1 = in_sizes[0] / (Bsz * D);
    int N2 = in_sizes[1] / (Bsz * D);

    chamfer_init_kernel<<<dim3(1), dim3(64), 0, stream>>>(out);

    dim3 blk(BLOCK);
    dim3 g1((N1 + ROWS_PER_WG - 1) / ROWS_PER_WG, Bsz);
    dim3 g2((N2 + ROWS_PER_WG - 1) / ROWS_PER_WG, Bsz);

    // dist1: rows from xyz1, candidates from xyz2
    chamfer_dir_kernel<<<g1, blk, 0, stream>>>(xyz1, xyz2, N1, N2,
                                               1.0f / ((float)Bsz * (float)N1), out);
    // dist2: rows from xyz2, candidates from xyz1
    chamfer_dir_kernel<<<g2, blk, 0, stream>>>(xyz2, xyz1, N2, N1,
                                               1.0f / ((float)Bsz * (float)N2), out);
}